// LocalGCN_84507776516613
// MI455X (gfx1250) — compile-verified
//
#include <hip/hip_runtime.h>
#include <math.h>

typedef __attribute__((ext_vector_type(2))) float v2f;
typedef __attribute__((ext_vector_type(8))) float v8f;

constexpr int TILES = 4;    // 16-row tiles per block, one per wave PAIR (8 waves total)
constexpr int WTILE = 16;   // rows per tile (WMMA M)
constexpr int LDW   = 65;   // padded LDS row stride
constexpr int LDX   = 13;   // padded stride for 12-wide staging tiles

__device__ __forceinline__ v8f v8f_zero() {
    v8f z = {0.f, 0.f, 0.f, 0.f, 0.f, 0.f, 0.f, 0.f};
    return z;
}

// V_WMMA_F32_16X16X4_F32 fragment layouts (CDNA5 ISA):
//   A frag (16x4):  VGPR r, lane l -> A[l&15][k0 + r + 2*(l>>4)]
//   B frag (4x16):  VGPR r, lane l -> B[k0 + r + 2*(l>>4)][n0 + (l&15)]
//   C/D (16x16):    VGPR v, lane l -> C[v + 8*(l>>4)][n0 + (l&15)]
template<int K4>  // K/4 fragments
__device__ __forceinline__ void load_a_frags(v2f* af, const float* A, int lda, int lane) {
    const float* p = A + (lane & 15) * lda + 2 * (lane >> 4);
    #pragma unroll
    for (int k = 0; k < K4; ++k) { af[k].x = p[4 * k]; af[k].y = p[4 * k + 1]; }
}

template<int K4>
__device__ __forceinline__ void load_b_frags(v2f* bf, const float* Bw, int ldb, int n0, int lane) {
    const float* p = Bw + (2 * (lane >> 4)) * ldb + n0 + (lane & 15);
    #pragma unroll
    for (int k = 0; k < K4; ++k) { bf[k].x = p[(4 * k) * ldb]; bf[k].y = p[(4 * k + 1) * ldb]; }
}

template<int K4>
__device__ __forceinline__ v8f wmma_mac(v8f acc, const v2f* af, const v2f* bf) {
    #pragma unroll
    for (int k = 0; k < K4; ++k)
        acc = __builtin_amdgcn_wmma_f32_16x16x4_f32(false, af[k], false, bf[k],
                                                    (short)0, acc, false, false);
    return acc;
}

__device__ __forceinline__ void store_d(float* D, int ldd, int n0, int lane,
                                        v8f acc, const float* bias) {
    const int n  = n0 + (lane & 15);
    const int mb = (lane >> 4) * 8;
    const float bv = bias[n];
    #pragma unroll
    for (int v = 0; v < 8; ++v) D[(mb + v) * ldd + n] = acc[v] + bv;
}

// LayerNorm(64)+ReLU over 8 rows of a [16][ldt] tile (rows half*8..half*8+7).
// 4 lanes per row, 16 columns each; reduce with shfl_xor(1),(2).
__device__ __forceinline__ void ln_relu_rows8(float* T, int ldt, const float* g, const float* be,
                                              int half, int lane, float* gout)
{
    const int r  = half * 8 + (lane >> 2);
    const int c0 = (lane & 3) * 16;
    float buf[16];
    #pragma unroll
    for (int j = 0; j < 16; ++j) buf[j] = T[r * ldt + c0 + j];
    float s = 0.f;
    #pragma unroll
    for (int j = 0; j < 16; ++j) s += buf[j];
    s += __shfl_xor(s, 1, 32);
    s += __shfl_xor(s, 2, 32);
    const float mu = s * (1.0f / 64.0f);
    float var = 0.f;
    #pragma unroll
    for (int j = 0; j < 16; ++j) { float d = buf[j] - mu; var += d * d; }
    var += __shfl_xor(var, 1, 32);
    var += __shfl_xor(var, 2, 32);
    const float inv = rsqrtf(var * (1.0f / 64.0f) + 1e-5f);
    if (gout) {
        float o[16];
        #pragma unroll
        for (int j = 0; j < 16; ++j)
            o[j] = fmaxf((buf[j] - mu) * inv * g[c0 + j] + be[c0 + j], 0.0f);
        float4* dst = (float4*)(gout + r * 64 + c0);  // 64B-aligned -> global_store_b128
        #pragma unroll
        for (int q = 0; q < 4; ++q)
            dst[q] = make_float4(o[4 * q], o[4 * q + 1], o[4 * q + 2], o[4 * q + 3]);
    } else {
        #pragma unroll
        for (int j = 0; j < 16; ++j)
            T[r * ldt + c0 + j] = fmaxf((buf[j] - mu) * inv * g[c0 + j] + be[c0 + j], 0.0f);
    }
}

__global__ __launch_bounds__(256)
void local_gcn_kernel(const float* __restrict__ xcur,   // [B,9]
                      const float* __restrict__ xnbr,   // [B,8,9]
                      const int*   __restrict__ mask,   // [B,8]
                      const float* __restrict__ W_enc, const float* __restrict__ b_enc,
                      const float* __restrict__ g_enc, const float* __restrict__ be_enc,
                      const float* __restrict__ W_a1,  const float* __restrict__ b_a1,
                      const float* __restrict__ W_a2,  const float* __restrict__ b_a2,
                      const float* __restrict__ W_out, const float* __restrict__ b_out,
                      const float* __restrict__ g_out, const float* __restrict__ be_out,
                      float* __restrict__ out, int B)
{
    // ---- block-shared weights (padded rows) ----
    __shared__ float sW_enc[12 * LDW];   // K padded 9 -> 12 with zero rows
    __shared__ float sW_a1 [128 * LDW];
    __shared__ float sW_out[128 * LDW];
    __shared__ float sW_a2[64];
    __shared__ float sb_enc[64], sg_enc[64], sbe_enc[64];
    __shared__ float sb_a1[64], sb_out[64], sg_out[64], sbe_out[64];
    __shared__ float s_ba2;

    // ---- per-tile working storage (one tile per wave pair) ----
    __shared__ float sX  [TILES][8][WTILE * LDX];   // staged raw inputs (slot 0 doubles for cur)
    __shared__ float sCur[TILES][WTILE * LDW];      // encoded current node
    __shared__ float sNbr[TILES][8][WTILE * LDW];   // encoded neighbors (slot 0 reused for out)
    __shared__ float sAgg[TILES][WTILE * LDW];      // attention-weighted aggregate
    __shared__ float sScP[TILES][2][8][WTILE];      // per-half partial scores [n][r]
    __shared__ float sSc [TILES][WTILE * 8];        // softmax weights [r][n]

    const int tid  = threadIdx.x;
    const int wave = tid >> 5;
    const int lane = tid & 31;
    const int pair = wave >> 1;     // tile index within block
    const int half = wave & 1;      // column-half owned by this wave
    const int pl   = half * 32 + lane;  // 0..63 within the pair

    // ---- cooperative weight / bias staging ----
    for (int idx = tid; idx < 12 * 64; idx += blockDim.x) {
        int r = idx >> 6, c = idx & 63;
        sW_enc[r * LDW + c] = (r < 9) ? W_enc[r * 64 + c] : 0.0f;
    }
    for (int idx = tid; idx < 128 * 64; idx += blockDim.x) {
        int r = idx >> 6, c = idx & 63;
        sW_a1 [r * LDW + c] = W_a1[idx];
        sW_out[r * LDW + c] = W_out[idx];
    }
    for (int idx = tid; idx < 64; idx += blockDim.x) {
        sW_a2 [idx] = W_a2[idx];
        sb_enc[idx] = b_enc[idx];  sg_enc[idx] = g_enc[idx];  sbe_enc[idx] = be_enc[idx];
        sb_a1 [idx] = b_a1[idx];   sb_out[idx] = b_out[idx];
        sg_out[idx] = g_out[idx];  sbe_out[idx] = be_out[idx];
    }
    if (tid == 0) s_ba2 = b_a2[0];
    __syncthreads();

    const int b0 = (blockIdx.x * TILES + pair) * WTILE;  // B = 131072 divides evenly

    float* myCur = sCur[pair];
    float* myAgg = sAgg[pair];

    // ===== stage current-node rows (9 cols -> 12, zero padded) =====
    {
        float* x0 = sX[pair][0];
        for (int idx = pl; idx < WTILE * 12; idx += 64) {
            int r = idx / 12, c = idx - 12 * r;
            x0[r * LDX + c] = (c < 9) ? xcur[(b0 + r) * 9 + c] : 0.0f;
        }
    }
    __syncthreads();

    // ===== encode current node: this wave covers n0 = half*32 + {0,16} =====
    {
        v2f af[3];
        load_a_frags<3>(af, sX[pair][0], LDX, lane);
        #pragma unroll
        for (int c = 0; c < 2; ++c) {
            const int n0 = half * 32 + c * 16;
            v2f bf[3];
            load_b_frags<3>(bf, sW_enc, LDW, n0, lane);
            v8f acc = v8f_zero();
            acc = wmma_mac<3>(acc, af, bf);
            store_d(myCur, LDW, n0, lane, acc, sb_enc);
        }
    }
    __syncthreads();

    // ===== LN+ReLU cur (half the rows each)  |  stage all 8 neighbor inputs =====
    ln_relu_rows8(myCur, LDW, sg_enc, sbe_enc, half, lane, nullptr);
    for (int idx = pl; idx < 8 * WTILE * 12; idx += 64) {
        int n = idx / 192, rem = idx - n * 192;
        int r = rem / 12, c = rem - 12 * r;
        sX[pair][n][r * LDX + c] = (c < 9) ? xnbr[((b0 + r) * 8 + n) * 9 + c] : 0.0f;
    }
    __syncthreads();

    // ===== encode neighbors (weight B-frags hoisted across all 8) =====
    {
        v2f bf0[3], bf1[3];
        load_b_frags<3>(bf0, sW_enc, LDW, half * 32,      lane);
        load_b_frags<3>(bf1, sW_enc, LDW, half * 32 + 16, lane);
        for (int n = 0; n < 8; ++n) {
            v2f af[3];
            load_a_frags<3>(af, sX[pair][n], LDX, lane);
            v8f acc0 = v8f_zero(), acc1 = v8f_zero();
            acc0 = wmma_mac<3>(acc0, af, bf0);
            acc1 = wmma_mac<3>(acc1, af, bf1);
            store_d(sNbr[pair][n], LDW, half * 32,      lane, acc0, sb_enc);
            store_d(sNbr[pair][n], LDW, half * 32 + 16, lane, acc1, sb_enc);
        }
    }
    __syncthreads();

    for (int n = 0; n < 8; ++n)
        ln_relu_rows8(sNbr[pair][n], LDW, sg_enc, sbe_enc, half, lane, nullptr);
    __syncthreads();

    // ===== attention scores, straight from WMMA accumulators =====
    // score[r] = sum_c relu(curA1[r][c] + nbrA1[r][c] + b_a1[c]) * W_a2[c]  (+ b_a2 later)
    const float ba2 = s_ba2;
    #pragma unroll
    for (int c = 0; c < 2; ++c) {
        const int n0   = half * 32 + c * 16;
        const int ncol = n0 + (lane & 15);
        const float ba1 = sb_a1[ncol];
        const float w2  = sW_a2[ncol];
        v2f af[16], bf[16];
        // cur @ W_a1[:64] chunk, kept in registers and shared by all 8 neighbors
        load_a_frags<16>(af, myCur, LDW, lane);
        load_b_frags<16>(bf, sW_a1, LDW, n0, lane);
        v8f base = v8f_zero();
        base = wmma_mac<16>(base, af, bf);
        // hoist W_a1[64:] fragments across the neighbor loop
        load_b_frags<16>(bf, sW_a1 + 64 * LDW, LDW, n0, lane);
        for (int n = 0; n < 8; ++n) {
            load_a_frags<16>(af, sNbr[pair][n], LDW, lane);
            v8f acc = v8f_zero();
            acc = wmma_mac<16>(acc, af, bf);
            float part[8];
            #pragma unroll
            for (int v = 0; v < 8; ++v)
                part[v] = fmaxf(acc[v] + base[v] + ba1, 0.0f) * w2;
            // reduce over the 16 columns held in lanes (l&15); lanes 0 and 16
            // end up with rows v and v+8 respectively
            #pragma unroll
            for (int v = 0; v < 8; ++v) {
                float p = part[v];
                p += __shfl_xor(p, 1, 32);
                p += __shfl_xor(p, 2, 32);
                p += __shfl_xor(p, 4, 32);
                p += __shfl_xor(p, 8, 32);
                part[v] = p;
            }
            if ((lane & 15) == 0) {
                float* dst = &sScP[pair][half][n][(lane >> 4) * 8];
                #pragma unroll
                for (int v = 0; v < 8; ++v)
                    dst[v] = (c == 0) ? part[v] : (dst[v] + part[v]);
            }
        }
    }
    __syncthreads();

    // ===== masked softmax over neighbors (one wave per tile, 16 lanes) =====
    if (half == 0 && lane < 16) {
        const int r = lane;
        const int* mrow = mask + (b0 + r) * 8;
        float sc[8], w[8]; int vld[8];
        float mx = -3.402823466e38f; int any = 0;
        #pragma unroll
        for (int n = 0; n < 8; ++n) {
            vld[n] = (mrow[n] != 0);
            sc[n] = sScP[pair][0][n][r] + sScP[pair][1][n][r] + ba2;
            if (vld[n]) { any = 1; mx = fmaxf(mx, sc[n]); }
        }
        float sum = 0.f;
        #pragma unroll
        for (int n = 0; n < 8; ++n) { w[n] = vld[n] ? expf(sc[n] - mx) : 0.0f; sum += w[n]; }
        const float invs = any ? (1.0f / sum) : 0.0f;   // all-masked row -> all-zero weights
        #pragma unroll
        for (int n = 0; n < 8; ++n) sSc[pair][r * 8 + n] = w[n] * invs;
    }
    __syncthreads();

    // ===== aggregate neighbors (this wave: its 32-column half) =====
    for (int idx = lane; idx < WTILE * 32; idx += 32) {
        const int r = idx >> 5, cc = (idx & 31) + half * 32;
        float s = 0.f;
        #pragma unroll
        for (int n = 0; n < 8; ++n) s += sSc[pair][r * 8 + n] * sNbr[pair][n][r * LDW + cc];
        myAgg[r * LDW + cc] = s;
    }
    __syncthreads();

    // ===== output: [cur, agg] @ W_out + b_out  -> reuse sNbr[pair][0] as scratch =====
    float* myOut = sNbr[pair][0];
    #pragma unroll
    for (int c = 0; c < 2; ++c) {
        const int n0 = half * 32 + c * 16;
        v2f af[16], bf[16];
        load_a_frags<16>(af, myCur, LDW, lane);
        load_b_frags<16>(bf, sW_out, LDW, n0, lane);
        v8f acc = v8f_zero();
        acc = wmma_mac<16>(acc, af, bf);
        load_a_frags<16>(af, myAgg, LDW, lane);
        load_b_frags<16>(bf, sW_out + 64 * LDW, LDW, n0, lane);
        acc = wmma_mac<16>(acc, af, bf);
        store_d(myOut, LDW, n0, lane, acc, sb_out);
    }
    __syncthreads();

    // ===== final LN+ReLU straight to global =====
    ln_relu_rows8(myOut, LDW, sg_out, sbe_out, half, lane, out + (size_t)b0 * 64);
}

extern "C" void kernel_launch(void* const* d_in, const int* in_sizes, int n_in,
                              void* d_out, int out_size, void* d_ws, size_t ws_size,
                              hipStream_t stream) {
    const float* xcur   = (const float*)d_in[0];
    const float* xnbr   = (const float*)d_in[1];
    const int*   mask   = (const int*)  d_in[2];
    const float* W_enc  = (const float*)d_in[3];
    const float* b_enc  = (const float*)d_in[4];
    const float* g_enc  = (const float*)d_in[5];
    const float* be_enc = (const float*)d_in[6];
    const float* W_a1   = (const float*)d_in[7];
    const float* b_a1   = (const float*)d_in[8];
    const float* W_a2   = (const float*)d_in[9];
    const float* b_a2   = (const float*)d_in[10];
    const float* W_out  = (const float*)d_in[11];
    const float* b_out  = (const float*)d_in[12];
    const float* g_out  = (const float*)d_in[13];
    const float* be_out = (const float*)d_in[14];
    float* out = (float*)d_out;

    const int B = in_sizes[0] / 9;                 // 131072
    const int rows_per_block = TILES * WTILE;      // 64
    const int blocks = (B + rows_per_block - 1) / rows_per_block;

    local_gcn_kernel<<<blocks, 256, 0, stream>>>(
        xcur, xnbr, mask,
        W_enc, b_enc, g_enc, be_enc,
        W_a1, b_a1, W_a2, b_a2,
        W_out, b_out, g_out, be_out,
        out, B);
}